// Model_70471823393007
// MI455X (gfx1250) — compile-verified
//
#include <hip/hip_runtime.h>
#include <cstdint>

typedef float v2f __attribute__((ext_vector_type(2)));
typedef float v8f __attribute__((ext_vector_type(8)));

#define NB         128
#define DE         16384      // d*e = 128*128
#define COLS       256        // columns per workgroup
#define NSLICES    (DE / COLS)
#define TILES_PW   2          // 16-col tiles per wave (8 waves * 2 * 16 = 256)
#define LDS_STRIDE 272        // 256 + 16 pad floats -> +2 rows == +32 banks
#define CHUNK      16
#define NCHUNK     (NB / CHUNK)

// csum(i) = sum_{k<i} min(seq_len - 64k, 64)  (exact, incl. partial/negative tail)
__device__ __forceinline__ float csumf(int i, int seq_len) {
    int K = seq_len / 64;
    if (K > i) K = i;
    if (K < 0) K = 0;
    float full = 64.0f * (float)K;
    int n = i - K;
    if (n <= 0) return full;
    float tri = 0.5f * (float)(K + i - 1) * (float)n;   // sum of k, k=K..i-1
    return full + (float)n * (float)seq_len - 64.0f * tri;
}

__global__ __launch_bounds__(256) void lightning_kv_scan_wmma(
    const float* __restrict__ s_rates,     // [h]
    const float* __restrict__ kv,          // [b,h,nb,d,e]
    const float* __restrict__ kv_hist,     // [b,h,d,e]
    const int*   __restrict__ seq_len_p,   // [1]
    float*       __restrict__ out)         // [b,h,nb+1,d,e]
{
    __shared__ float Y[CHUNK * LDS_STRIDE];

    const int tid   = threadIdx.x;
    const int lane  = tid & 31;
    const int wave  = tid >> 5;
    const int n15   = lane & 15;
    const int hi    = lane >> 4;            // 0 = lanes 0-15, 1 = lanes 16-31
    const int slice = blockIdx.x;           // 0..NSLICES-1
    const int bh    = blockIdx.y;           // 0..31
    const int h     = bh & 15;

    const int   SL = *seq_len_p;
    const float sh = s_rates[h];

    const size_t kv_bh  = (size_t)bh * NB * DE;
    const size_t out_bh = (size_t)bh * (NB + 1) * DE;
    const size_t col0   = (size_t)slice * COLS;
    const int    tbase0 = wave * TILES_PW * 16;

    // Carry state S = state_0 = kv_history (one column per lane per tile,
    // replicated across lane halves). Also emit output row 0 (exclusive state).
    float S[TILES_PW];
    for (int t = 0; t < TILES_PW; ++t) {
        const int cb = tbase0 + t * 16 + n15;
        S[t] = kv_hist[(size_t)bh * DE + col0 + cb];
        if (hi == 0)
            __builtin_nontemporal_store(S[t], &out[out_bh + col0 + cb]);
    }

    // Low 32 bits of the flat shared address == LDS byte offset (ISA aperture rule).
    const uint32_t lds_base = (uint32_t)(uintptr_t)(&Y[0]);
    const int ld_row = tid >> 4;   // 0..15 : chunk-local block row
    const int ld_c4  = tid & 15;   // float4 column group

    for (int c = 0; c < NCHUNK; ++c) {
        __syncthreads();   // previous chunk's LDS reads complete before overwrite

        // ---- async stage: 16 rows x 256 floats, 4x b128 per thread ----
        for (int j = 0; j < 4; ++j) {
            const int col4 = ld_c4 + 16 * j;
            const float* gp = kv + kv_bh + (size_t)(c * CHUNK + ld_row) * DE
                                 + col0 + (size_t)col4 * 4;
            const uint32_t lp = lds_base
                              + (uint32_t)((ld_row * LDS_STRIDE + col4 * 4) * 4);
            asm volatile("global_load_async_to_lds_b128 %0, %1, off"
                         :: "v"(lp), "v"((uint64_t)(uintptr_t)gp) : "memory");
        }
        asm volatile("s_wait_asynccnt 0" ::: "memory");
        __syncthreads();   // all waves' async transfers visible

        // ---- per-chunk decay coefficients (uniform ALU, 16 expf/lane) ----
        const float cs_base = csumf(16 * c, SL);
        const int   r       = n15;                       // A-matrix row (M)
        const float cs_r1   = csumf(16 * c + r + 1, SL);

        v2f A4[4];   // A[r][k] laid out per ISA: VGPR0 = K=k0 / k0+2, VGPR1 = +1
        for (int ks = 0; ks < 4; ++ks) {
            const int m0 = ks * 4 + hi * 2;
            A4[ks].x = (m0     <= r)
                ? __expf(-sh * (cs_r1 - csumf(16 * c + m0 + 1, SL))) : 0.0f;
            A4[ks].y = (m0 + 1 <= r)
                ? __expf(-sh * (cs_r1 - csumf(16 * c + m0 + 2, SL))) : 0.0f;
        }
        float alpha[8];  // decay applied to carry S for D row v (lane-half selected)
        for (int v = 0; v < 8; ++v) {
            const int rr = v + hi * 8;
            alpha[v] = __expf(-sh * (csumf(16 * c + rr + 1, SL) - cs_base));
        }

        // ---- triangular matmul per 16-col tile ----
        for (int t = 0; t < TILES_PW; ++t) {
            const int cb = tbase0 + t * 16 + n15;
            v8f D = {0.f, 0.f, 0.f, 0.f, 0.f, 0.f, 0.f, 0.f};
            #pragma unroll
            for (int ks = 0; ks < 4; ++ks) {
                const int kk = ks * 4 + hi * 2;  // B: VGPR0 = K=kk, VGPR1 = K=kk+1
                v2f B;
                B.x = Y[kk * LDS_STRIDE + cb];
                B.y = Y[(kk + 1) * LDS_STRIDE + cb];
                D = __builtin_amdgcn_wmma_f32_16x16x4_f32(
                        /*neg_a=*/false, A4[ks], /*neg_b=*/false, B,
                        /*c_mod=*/(short)0, D, /*reuse_a=*/false, /*reuse_b=*/false);
            }
            #pragma unroll
            for (int v = 0; v < 8; ++v) D[v] += alpha[v] * S[t];

            // store states 16c+1 .. 16c+16 (rows v / v+8 per lane half)
            #pragma unroll
            for (int v = 0; v < 8; ++v) {
                const int row = 16 * c + 1 + v + hi * 8;
                __builtin_nontemporal_store(
                    D[v], &out[out_bh + (size_t)row * DE + col0 + cb]);
            }
            // carry = state_{16c+16} = matmul row 15 (VGPR7, upper lanes)
            S[t] = __shfl(D[7], 16 + n15, 32);
        }
    }
}

extern "C" void kernel_launch(void* const* d_in, const int* in_sizes, int n_in,
                              void* d_out, int out_size, void* d_ws, size_t ws_size,
                              hipStream_t stream) {
    (void)in_sizes; (void)n_in; (void)out_size; (void)d_ws; (void)ws_size;
    const float* s_rates = (const float*)d_in[0];
    const float* kv      = (const float*)d_in[1];
    const float* kv_hist = (const float*)d_in[2];
    const int*   seq_len = (const int*)d_in[3];
    float* out = (float*)d_out;

    dim3 grid(NSLICES, 2 * 16);   // (column slices, b*h)
    lightning_kv_scan_wmma<<<grid, 256, 0, stream>>>(s_rates, kv, kv_hist, seq_len, out);
}